// Attention_24610162606157
// MI455X (gfx1250) — compile-verified
//
#include <hip/hip_runtime.h>
#include <math.h>

typedef __attribute__((ext_vector_type(2))) float v2f;
typedef __attribute__((ext_vector_type(8))) float v8f;

#define B_DIM 4096
#define L_DIM 64
#define D_EMB 256
#define H_DIM 20
#define R_ROWS (B_DIM * L_DIM)          // 262144 rows of [256]
#define PROJ_BLOCKS (R_ROWS / 128)      // 2048 (8 waves x 16 rows per block)

__device__ __forceinline__ v8f wmma4(v2f a, v2f b, v8f c) {
  // D = A(16x4,f32) * B(4x16,f32) + C(16x16,f32)
  return __builtin_amdgcn_wmma_f32_16x16x4_f32(false, a, false, b, (short)0, c,
                                               false, false);
}

// ---------------------------------------------------------------------------
// Pass 1: projections  Y[r,0..19] = X[r,:] @ W[256,20] + bias, for 16-row tile
// ---------------------------------------------------------------------------
__device__ __forceinline__ void proj16(const float* __restrict__ X,
                                       const float* __restrict__ W,
                                       const float* __restrict__ bias,
                                       float* __restrict__ Y, long r0, int lmod,
                                       int lhalf, float* sum, float* ssq) {
  const float2* xp = (const float2*)(X + (r0 + lmod) * D_EMB + lhalf * 2);
#pragma unroll
  for (int c0 = 0; c0 < 32; c0 += 16) {
    int col = c0 + lmod;
    v8f c = {};
    for (int k0 = 0; k0 < D_EMB; k0 += 4) {
      int ka = k0 + lhalf * 2;
      float2 av = xp[k0 >> 1];
      v2f a; a.x = av.x; a.y = av.y;
      v2f b;
      if (col < H_DIM) {
        b.x = W[ka * H_DIM + col];
        b.y = W[(ka + 1) * H_DIM + col];
      } else {
        b.x = 0.f; b.y = 0.f;
      }
      c = wmma4(a, b, c);
    }
    if (col < H_DIM) {
      float bb = bias[col];
#pragma unroll
      for (int j = 0; j < 8; j++) {
        float v = c[j] + bb;
        Y[(r0 + lhalf * 8 + j) * H_DIM + col] = v;
        if (sum) { *sum += v; *ssq += v * v; }
      }
    }
  }
}

__global__ void proj_kernel(const float* __restrict__ key,
                            const float* __restrict__ query,
                            const float* __restrict__ value,
                            const float* __restrict__ Wk, const float* __restrict__ bk,
                            const float* __restrict__ Wq, const float* __restrict__ bq,
                            const float* __restrict__ Wv, const float* __restrict__ bv,
                            float* __restrict__ Qp, float* __restrict__ Kp,
                            float* __restrict__ Vp, float* __restrict__ partials) {
  int tid = threadIdx.x;
  int wave = tid >> 5, lane = tid & 31;
  int lmod = lane & 15, lhalf = lane >> 4;
  long r0 = (long)blockIdx.x * 128 + wave * 16;

  float sq = 0.f, ssq = 0.f, sk = 0.f, ssk = 0.f;
  proj16(query, Wq, bq, Qp, r0, lmod, lhalf, &sq, &ssq);
  proj16(key,   Wk, bk, Kp, r0, lmod, lhalf, &sk, &ssk);
  proj16(value, Wv, bv, Vp, r0, lmod, lhalf, nullptr, nullptr);

  __shared__ float4 red[256];
  red[tid] = make_float4(sq, ssq, sk, ssk);
  __syncthreads();
  for (int s2 = 128; s2 > 0; s2 >>= 1) {
    if (tid < s2) {
      red[tid].x += red[tid + s2].x;
      red[tid].y += red[tid + s2].y;
      red[tid].z += red[tid + s2].z;
      red[tid].w += red[tid + s2].w;
    }
    __syncthreads();
  }
  if (tid == 0) ((float4*)partials)[blockIdx.x] = red[0];
}

// ---------------------------------------------------------------------------
// Pass 2: per-l reductions over n. Augmented matrices:
//   A = [K_l^T ; ones ; 0] (32 x n),  B = [V_l , ones , 0] (n x 32)
//   C[0:20,0:20]=kvs[l], C[m,20]=ks_sum[l][m], C[20,d]=vs_sum[l][d]
// ---------------------------------------------------------------------------
__global__ void kvs_kernel(const float* __restrict__ Kp,
                           const float* __restrict__ Vp,
                           float* __restrict__ kvs, float* __restrict__ ksum,
                           float* __restrict__ vsum) {
  int l = blockIdx.x;
  int tid = threadIdx.x;
  int wave = tid >> 5, lane = tid & 31;
  int lmod = lane & 15, lhalf = lane >> 4;

  v8f c00 = {}, c01 = {}, c10 = {}, c11 = {};
  int nbase = wave * (B_DIM / 8);   // 512 n's per wave
  int m1 = 16 + lmod;
  int d1 = 16 + lmod;

  for (int kk = 0; kk < B_DIM / 8; kk += 4) {
    int n = nbase + kk + lhalf * 2;
    long row0 = ((long)n * L_DIM + l) * H_DIM;
    long row1 = ((long)(n + 1) * L_DIM + l) * H_DIM;

    v2f a0; a0.x = Kp[row0 + lmod]; a0.y = Kp[row1 + lmod];
    v2f a1;
    a1.x = (m1 < H_DIM) ? Kp[row0 + m1] : ((m1 == H_DIM) ? 1.f : 0.f);
    a1.y = (m1 < H_DIM) ? Kp[row1 + m1] : ((m1 == H_DIM) ? 1.f : 0.f);

    v2f b0; b0.x = Vp[row0 + lmod]; b0.y = Vp[row1 + lmod];
    v2f b1;
    b1.x = (d1 < H_DIM) ? Vp[row0 + d1] : ((d1 == H_DIM) ? 1.f : 0.f);
    b1.y = (d1 < H_DIM) ? Vp[row1 + d1] : ((d1 == H_DIM) ? 1.f : 0.f);

    c00 = wmma4(a0, b0, c00);
    c01 = wmma4(a0, b1, c01);
    c10 = wmma4(a1, b0, c10);
    c11 = wmma4(a1, b1, c11);
  }

  __shared__ float part[8][32][32];
#pragma unroll
  for (int j = 0; j < 8; j++) {
    int r_lo = lhalf * 8 + j;
    part[wave][r_lo][lmod]           = c00[j];
    part[wave][r_lo][16 + lmod]      = c01[j];
    part[wave][16 + r_lo][lmod]      = c10[j];
    part[wave][16 + r_lo][16 + lmod] = c11[j];
  }
  __syncthreads();

  for (int idx = tid; idx < 1024; idx += 256) {
    int row = idx >> 5, col = idx & 31;
    float s = 0.f;
#pragma unroll
    for (int w = 0; w < 8; w++) s += part[w][row][col];
    if (row < H_DIM && col < H_DIM)
      kvs[l * (H_DIM * H_DIM) + row * H_DIM + col] = s;
    else if (row < H_DIM && col == H_DIM)
      ksum[l * H_DIM + row] = s;
    else if (row == H_DIM && col < H_DIM)
      vsum[l * H_DIM + col] = s;
  }
}

// ---------------------------------------------------------------------------
// Pass 3: deterministic reduction of block partials -> combined scale s
// ---------------------------------------------------------------------------
__global__ void scale_kernel(const float* __restrict__ partials,
                             float* __restrict__ scl) {
  __shared__ float4 red[256];
  int tid = threadIdx.x;
  float4 acc = make_float4(0.f, 0.f, 0.f, 0.f);
  for (int i = tid; i < PROJ_BLOCKS; i += 256) {
    float4 p = ((const float4*)partials)[i];
    acc.x += p.x; acc.y += p.y; acc.z += p.z; acc.w += p.w;
  }
  red[tid] = acc;
  __syncthreads();
  for (int s2 = 128; s2 > 0; s2 >>= 1) {
    if (tid < s2) {
      red[tid].x += red[tid + s2].x;
      red[tid].y += red[tid + s2].y;
      red[tid].z += red[tid + s2].z;
      red[tid].w += red[tid + s2].w;
    }
    __syncthreads();
  }
  if (tid == 0) {
    float sumq = red[0].x, ssq = red[0].y, sumk = red[0].z, ssk = red[0].w;
    bool cond = (sumq != 0.f) && (sumk != 0.f);
    scl[0] = cond ? (1.0f / (sqrtf(ssq) * sqrtf(ssk))) : 1.0f;
  }
}

// ---------------------------------------------------------------------------
// Pass 4: attn = (s * qs@kvs[l] + vs_sum[l]) / (s * qs.ks_sum[l] + B)
//         out = attn @ Wp + bp        (WMMA over K=20, N=256)
// Rows in a 16-row tile share l (n varies) so kvs[l] is tile-uniform.
// ---------------------------------------------------------------------------
__global__ void attn_out_kernel(const float* __restrict__ Qp,
                                const float* __restrict__ kvs,
                                const float* __restrict__ ksum,
                                const float* __restrict__ vsum,
                                const float* __restrict__ scl,
                                const float* __restrict__ Wp,
                                const float* __restrict__ bp,
                                float* __restrict__ out) {
  int l = blockIdx.x;
  int chunk = blockIdx.y;
  int tid = threadIdx.x;
  int wave = tid >> 5, lane = tid & 31;
  int lmod = lane & 15, lhalf = lane >> 4;

  __shared__ float kvsL[H_DIM][H_DIM];
  __shared__ float ksumL[H_DIM];
  __shared__ float vsumL[H_DIM];
  __shared__ float attnL[8][16][H_DIM];

  for (int i = tid; i < H_DIM * H_DIM; i += 256)
    kvsL[i / H_DIM][i % H_DIM] = kvs[l * (H_DIM * H_DIM) + i];
  if (tid < H_DIM) {
    ksumL[tid] = ksum[l * H_DIM + tid];
    vsumL[tid] = vsum[l * H_DIM + tid];
  }
  float s = scl[0];
  __syncthreads();

  int n0 = chunk * 128 + wave * 16;

  if (lane < 16) {
    long r = (long)(n0 + lane) * L_DIM + l;
    const float4* qv = (const float4*)(Qp + r * H_DIM);
    float q[H_DIM];
#pragma unroll
    for (int m4 = 0; m4 < 5; m4++) {
      float4 v = qv[m4];
      q[m4 * 4 + 0] = v.x; q[m4 * 4 + 1] = v.y;
      q[m4 * 4 + 2] = v.z; q[m4 * 4 + 3] = v.w;
    }
    float den = 0.f;
#pragma unroll
    for (int m = 0; m < H_DIM; m++) den += q[m] * ksumL[m];
    float inv = 1.0f / (s * den + (float)B_DIM);
#pragma unroll
    for (int d = 0; d < H_DIM; d++) {
      float num = 0.f;
#pragma unroll
      for (int m = 0; m < H_DIM; m++) num += q[m] * kvsL[m][d];
      attnL[wave][lane][d] = (s * num + vsumL[d]) * inv;
    }
  }
  __syncthreads();

  // out[16 x 256] = attn[16 x 20] @ Wp[20 x 256] + bp
  for (int c0 = 0; c0 < D_EMB; c0 += 16) {
    int col = c0 + lmod;
    v8f c = {};
#pragma unroll
    for (int k0 = 0; k0 < H_DIM; k0 += 4) {
      int ka = k0 + lhalf * 2;
      v2f a; a.x = attnL[wave][lmod][ka]; a.y = attnL[wave][lmod][ka + 1];
      v2f b; b.x = Wp[ka * D_EMB + col]; b.y = Wp[(ka + 1) * D_EMB + col];
      c = wmma4(a, b, c);
    }
    float bb = bp[col];
#pragma unroll
    for (int j = 0; j < 8; j++) {
      int m = lhalf * 8 + j;
      long r = (long)(n0 + m) * L_DIM + l;
      out[r * D_EMB + col] = c[j] + bb;
    }
  }
}

// ---------------------------------------------------------------------------
extern "C" void kernel_launch(void* const* d_in, const int* in_sizes, int n_in,
                              void* d_out, int out_size, void* d_ws,
                              size_t ws_size, hipStream_t stream) {
  (void)in_sizes; (void)n_in; (void)out_size; (void)ws_size;

  const float* key   = (const float*)d_in[0];
  const float* value = (const float*)d_in[1];
  const float* query = (const float*)d_in[2];
  const float* Wk = (const float*)d_in[3];
  const float* bk = (const float*)d_in[4];
  const float* Wq = (const float*)d_in[5];
  const float* bq = (const float*)d_in[6];
  const float* Wv = (const float*)d_in[7];
  const float* bv = (const float*)d_in[8];
  const float* Wp = (const float*)d_in[9];
  const float* bp = (const float*)d_in[10];

  float* ws = (float*)d_ws;
  const size_t PROJ_ELEMS = (size_t)R_ROWS * H_DIM;  // 5,242,880
  float* Qp = ws;
  float* Kp = ws + PROJ_ELEMS;
  float* Vp = ws + 2 * PROJ_ELEMS;
  float* kvs      = ws + 3 * PROJ_ELEMS;                      // 64*400
  float* ksum     = kvs + L_DIM * H_DIM * H_DIM;              // 64*20
  float* vsum     = ksum + L_DIM * H_DIM;                     // 64*20
  float* partials = vsum + L_DIM * H_DIM;                     // 2048*4
  float* scl      = partials + PROJ_BLOCKS * 4;               // 1

  proj_kernel<<<PROJ_BLOCKS, 256, 0, stream>>>(key, query, value, Wk, bk, Wq,
                                               bq, Wv, bv, Qp, Kp, Vp, partials);
  kvs_kernel<<<L_DIM, 256, 0, stream>>>(Kp, Vp, kvs, ksum, vsum);
  scale_kernel<<<1, 256, 0, stream>>>(partials, scl);
  attn_out_kernel<<<dim3(L_DIM, 32), 256, 0, stream>>>(
      Qp, kvs, ksum, vsum, scl, Wp, bp, (float*)d_out);
}